// CLFM_70119636075167
// MI455X (gfx1250) — compile-verified
//
#include <hip/hip_runtime.h>
#include <math.h>

// ---------------------------------------------------------------------------
// Problem constants (from reference setup_inputs)
// ---------------------------------------------------------------------------
#define BB 2
#define NN 8192
#define HH 80
#define WW 160
#define HWN (HH * WW)      // 12800
#define C2 128
#define C3 64
#define KNN 3

typedef __attribute__((ext_vector_type(2))) float v2f;
typedef __attribute__((ext_vector_type(8))) float v8f;

__device__ __forceinline__ float lrelu(float x) { return (x >= 0.f) ? x : 0.1f * x; }
__device__ __forceinline__ float sigmoidf(float x) { return 1.0f / (1.0f + __expf(-x)); }

// ---------------------------------------------------------------------------
// Kernel 1: brute-force KNN (top-3 by squared distance), LDS-tiled.
// The uv tile is staged into LDS with gfx1250 async global->LDS copies
// (GLOBAL_LOAD_ASYNC_TO_LDS_B32, tracked by ASYNCcnt), then consumed after
// s_wait_asynccnt + barrier.
// grid: B * (HW/256) blocks of 256 threads; one query per thread
// ---------------------------------------------------------------------------
#define KNN_TILE 1024

__global__ __launch_bounds__(256) void knn_kernel(const float* __restrict__ uv,
                                                  int* __restrict__ idx_out) {
  __shared__ float sx[KNN_TILE];
  __shared__ float sy[KNN_TILE];

  const int blocks_per_batch = HWN / 256;            // 50
  const int b = blockIdx.x / blocks_per_batch;
  const int q = (blockIdx.x % blocks_per_batch) * 256 + threadIdx.x;

  const float gx = (float)(q % WW);
  const float gy = (float)(q / WW);

  const float* __restrict__ ux = uv + (size_t)b * 2 * NN;
  const float* __restrict__ uy = ux + NN;

  float d0 = 3.0e38f, d1 = 3.0e38f, d2 = 3.0e38f;
  int i0 = 0, i1 = 0, i2 = 0;

  for (int t = 0; t < NN; t += KNN_TILE) {
    __syncthreads();
    for (int j = threadIdx.x; j < KNN_TILE; j += 256) {
      // async copy global -> LDS (per-lane 4B), ASYNCcnt-tracked
      unsigned lx = (unsigned)(uintptr_t)&sx[j];
      unsigned ly = (unsigned)(uintptr_t)&sy[j];
      unsigned long long gax = (unsigned long long)(uintptr_t)(ux + t + j);
      unsigned long long gay = (unsigned long long)(uintptr_t)(uy + t + j);
      asm volatile("global_load_async_to_lds_b32 %0, %1, off"
                   :: "v"(lx), "v"(gax) : "memory");
      asm volatile("global_load_async_to_lds_b32 %0, %1, off"
                   :: "v"(ly), "v"(gay) : "memory");
    }
    asm volatile("s_wait_asynccnt 0x0" ::: "memory");
    __syncthreads();

    for (int j = 0; j < KNN_TILE; ++j) {
      const float dx = sx[j] - gx;
      const float dy = sy[j] - gy;
      const float d = dx * dx + dy * dy;
      const int n = t + j;
      if (d < d2) {
        if (d < d1) {
          d2 = d1; i2 = i1;
          if (d < d0) { d1 = d0; i1 = i0; d0 = d; i0 = n; }
          else        { d1 = d;  i1 = n; }
        } else {
          d2 = d; i2 = n;
        }
      }
    }
  }

  const int base = (b * HWN + q) * KNN;
  idx_out[base + 0] = i0;
  idx_out[base + 1] = i1;
  idx_out[base + 2] = i2;
}

// ---------------------------------------------------------------------------
// Kernel 2: score MLP (3 -> 3 -> 64, lrelu/sigmoid) + gathered weighted sum
// final[b, c, q] = sum_k sigmoid(w_s2 @ lrelu(w_s1 @ s_in + b_s1) + b_s2)[c]
//                  * feat3d[b, c, idx_k]
// one thread per (b, q)
// ---------------------------------------------------------------------------
__global__ __launch_bounds__(256) void interp_score_kernel(
    const float* __restrict__ uv, const float* __restrict__ feat3d,
    const int* __restrict__ knn_idx,
    const float* __restrict__ w_s1, const float* __restrict__ b_s1,
    const float* __restrict__ w_s2, const float* __restrict__ b_s2,
    float* __restrict__ final_out) {
  const int t = blockIdx.x * blockDim.x + threadIdx.x;
  if (t >= BB * HWN) return;
  const int b = t / HWN;
  const int q = t - b * HWN;

  const float gx = (float)(q % WW);
  const float gy = (float)(q / WW);

  const float* __restrict__ ux = uv + (size_t)b * 2 * NN;
  const float* __restrict__ uy = ux + NN;

  int nidx[KNN];
  float h[KNN][3];
  const int base = (b * HWN + q) * KNN;
#pragma unroll
  for (int k = 0; k < KNN; ++k) {
    const int n = knn_idx[base + k];
    nidx[k] = n;
    const float ox = ux[n] - gx;
    const float oy = uy[n] - gy;
    const float on = sqrtf(ox * ox + oy * oy);
#pragma unroll
    for (int j = 0; j < 3; ++j) {
      h[k][j] = lrelu(w_s1[j * 3 + 0] * ox + w_s1[j * 3 + 1] * oy +
                      w_s1[j * 3 + 2] * on + b_s1[j]);
    }
  }

  const float* __restrict__ f3 = feat3d + (size_t)b * C3 * NN;
  float* __restrict__ fo = final_out + (size_t)b * C3 * HWN + q;
  for (int c = 0; c < C3; ++c) {
    const float w0 = w_s2[c * 3 + 0], w1 = w_s2[c * 3 + 1], w2 = w_s2[c * 3 + 2];
    const float bc = b_s2[c];
    float acc = 0.f;
#pragma unroll
    for (int k = 0; k < KNN; ++k) {
      const float s = sigmoidf(w0 * h[k][0] + w1 * h[k][1] + w2 * h[k][2] + bc);
      acc += s * f3[(size_t)c * NN + nidx[k]];
    }
    fo[(size_t)c * HWN] = acc;
  }
}

// ---------------------------------------------------------------------------
// Kernel 3: pointwise conv as full-precision WMMA GEMM.
// Y[b,o,p] = act( sum_c Wm[o,c] * X[b,c,p] + bias[o] )
// One wave32 per 16x16 output tile; K stepped by 4 via V_WMMA_F32_16X16X4_F32.
// C and P are compile-time so the unrolled loop uses immediate-offset
// global loads (no per-iteration 64-bit address VALU).
// A (16x4 f32) lane layout: lanes 0-15 hold K,K+1 ; lanes 16-31 hold K+2,K+3.
// C/D (16x16 f32): VGPR r -> rows r (lanes 0-15) and r+8 (lanes 16-31).
// ---------------------------------------------------------------------------
template <int C, int P, int ACT>
__global__ __launch_bounds__(32) void gemm_pw(const float* __restrict__ Wm,
                                              const float* __restrict__ X,
                                              const float* __restrict__ bias,
                                              float* __restrict__ Y, int O) {
  const int lane = threadIdx.x;
  const int pt = blockIdx.x;   // pixel tile
  const int ot = blockIdx.y;   // output-channel tile
  const int b  = blockIdx.z;

  const int col  = pt * 16 + (lane & 15);
  const int rowA = ot * 16 + (lane & 15);
  const int h = lane >> 4;  // lane half selects K pair

  // Base pointers hoisted: all loop offsets are compile-time immediates.
  const float* __restrict__ wrow = Wm + (size_t)rowA * C + 2 * h;
  const float* __restrict__ xc =
      X + (size_t)b * C * P + (size_t)(2 * h) * P + col;

  v8f acc = {};
#pragma unroll
  for (int kk = 0; kk < C; kk += 4) {
    const v2f av = *(const v2f*)(wrow + kk);   // W[row, kk+2h], W[row, kk+2h+1]
    v2f bv;
    bv.x = xc[(size_t)kk * P];                 // X[kk+2h,   col]
    bv.y = xc[(size_t)kk * P + P];             // X[kk+2h+1, col]
    acc = __builtin_amdgcn_wmma_f32_16x16x4_f32(false, av, false, bv,
                                                (short)0, acc, false, false);
  }

  float* __restrict__ yb =
      Y + (size_t)b * O * P + (size_t)(ot * 16 + 8 * h) * P + col;
  const float* __restrict__ bb = bias + ot * 16 + 8 * h;
#pragma unroll
  for (int r = 0; r < 8; ++r) {
    float y = acc[r] + bb[r];
    if (ACT) y = lrelu(y);
    yb[(size_t)r * P] = y;
  }
}

// ---------------------------------------------------------------------------
// Kernel 4: bilinear grid sample  samp[b,c,n]  — one thread per (b,n)
// ---------------------------------------------------------------------------
__global__ __launch_bounds__(256) void grid_sample_kernel(
    const float* __restrict__ feat2d, const float* __restrict__ uv,
    float* __restrict__ samp) {
  const int t = blockIdx.x * blockDim.x + threadIdx.x;
  if (t >= BB * NN) return;
  const int b = t / NN;
  const int n = t - b * NN;

  const float x = uv[(size_t)b * 2 * NN + n];
  const float y = uv[(size_t)b * 2 * NN + NN + n];
  const float x0 = floorf(x), y0 = floorf(y);
  const float wx = x - x0, wy = y - y0;
  int x0i = (int)x0; x0i = x0i < 0 ? 0 : (x0i > WW - 1 ? WW - 1 : x0i);
  int y0i = (int)y0; y0i = y0i < 0 ? 0 : (y0i > HH - 1 ? HH - 1 : y0i);
  const int x1i = x0i + 1 > WW - 1 ? WW - 1 : x0i + 1;
  const int y1i = y0i + 1 > HH - 1 ? HH - 1 : y0i + 1;

  const float w00 = (1.f - wx) * (1.f - wy);
  const float w01 = wx * (1.f - wy);
  const float w10 = (1.f - wx) * wy;
  const float w11 = wx * wy;
  const int o00 = y0i * WW + x0i, o01 = y0i * WW + x1i;
  const int o10 = y1i * WW + x0i, o11 = y1i * WW + x1i;

  const float* __restrict__ f = feat2d + (size_t)b * C2 * HWN;
  float* __restrict__ s = samp + (size_t)b * C2 * NN + n;
#pragma unroll 4
  for (int c = 0; c < C2; ++c) {
    const float* fc = f + (size_t)c * HWN;
    s[(size_t)c * NN] = fc[o00] * w00 + fc[o01] * w01 + fc[o10] * w10 + fc[o11] * w11;
  }
}

// ---------------------------------------------------------------------------
// Kernel 5: pooled[b,c] = mean_p (A[b,c,p] + B[b,c,p]); one block per (b,c)
// ---------------------------------------------------------------------------
__global__ __launch_bounds__(256) void pool_mean(const float* __restrict__ A,
                                                 const float* __restrict__ Bv,
                                                 float* __restrict__ pooled, int P) {
  const int bc = blockIdx.x;
  const float* a = A + (size_t)bc * P;
  const float* bb = Bv + (size_t)bc * P;
  float s = 0.f;
  for (int p = threadIdx.x; p < P; p += 256) s += a[p] + bb[p];
  __shared__ float red[256];
  red[threadIdx.x] = s;
  __syncthreads();
  for (int st = 128; st > 0; st >>= 1) {
    if (threadIdx.x < st) red[threadIdx.x] += red[threadIdx.x + st];
    __syncthreads();
  }
  if (threadIdx.x == 0) pooled[bc] = red[0] / (float)P;
}

// ---------------------------------------------------------------------------
// Kernel 6: SK gate FC: mid = relu(pooled @ wmid^T); v = sigmoid(mid @ wout^T);
// softmax over pairs -> wsel[b, c, 0/1]. Single block, tiny sizes.
// ---------------------------------------------------------------------------
__global__ __launch_bounds__(256) void sk_fc(const float* __restrict__ pooled,
                                             const float* __restrict__ wmid,
                                             const float* __restrict__ wout,
                                             float* __restrict__ wsel,
                                             int Cout, int Cmid) {
  __shared__ float pooled_s[128];
  __shared__ float mid_s[64];
  __shared__ float v_s[256];
  const int tid = threadIdx.x;
  for (int b = 0; b < BB; ++b) {
    if (tid < Cout) pooled_s[tid] = pooled[b * Cout + tid];
    __syncthreads();
    if (tid < Cmid) {
      float m = 0.f;
      for (int c = 0; c < Cout; ++c) m += pooled_s[c] * wmid[tid * Cout + c];
      mid_s[tid] = m > 0.f ? m : 0.f;
    }
    __syncthreads();
    if (tid < 2 * Cout) {
      float v = 0.f;
      for (int m = 0; m < Cmid; ++m) v += mid_s[m] * wout[tid * Cmid + m];
      v_s[tid] = sigmoidf(v);
    }
    __syncthreads();
    if (tid < Cout) {
      const float e0 = __expf(v_s[2 * tid + 0]);
      const float e1 = __expf(v_s[2 * tid + 1]);
      const float inv = 1.0f / (e0 + e1);
      wsel[(b * Cout + tid) * 2 + 0] = e0 * inv;
      wsel[(b * Cout + tid) * 2 + 1] = e1 * inv;
    }
    __syncthreads();
  }
}

// ---------------------------------------------------------------------------
// Kernel 7: out[b,c,p] = A[b,c,p]*wsel[b,c,0] + B[b,c,p]*wsel[b,c,1]
// ---------------------------------------------------------------------------
__global__ __launch_bounds__(256) void sk_combine(const float* __restrict__ A,
                                                  const float* __restrict__ Bv,
                                                  const float* __restrict__ wsel,
                                                  float* __restrict__ out,
                                                  int P, int total) {
  const int t = blockIdx.x * blockDim.x + threadIdx.x;
  if (t >= total) return;
  const int bc = t / P;
  out[t] = A[t] * wsel[bc * 2 + 0] + Bv[t] * wsel[bc * 2 + 1];
}

// ---------------------------------------------------------------------------
// Host launcher
// ---------------------------------------------------------------------------
extern "C" void kernel_launch(void* const* d_in, const int* in_sizes, int n_in,
                              void* d_out, int out_size, void* d_ws, size_t ws_size,
                              hipStream_t stream) {
  const float* uv      = (const float*)d_in[0];
  const float* feat_2d = (const float*)d_in[1];
  const float* feat_3d = (const float*)d_in[2];
  const float* w_oc  = (const float*)d_in[3];
  const float* b_oc  = (const float*)d_in[4];
  const float* w_s1  = (const float*)d_in[5];
  const float* b_s1  = (const float*)d_in[6];
  const float* w_s2  = (const float*)d_in[7];
  const float* b_s2  = (const float*)d_in[8];
  const float* w_m3  = (const float*)d_in[9];
  const float* b_m3  = (const float*)d_in[10];
  const float* w2a1  = (const float*)d_in[11];
  const float* b2a1  = (const float*)d_in[12];
  const float* w2a2  = (const float*)d_in[13];
  const float* b2a2  = (const float*)d_in[14];
  const float* w2mid = (const float*)d_in[15];
  const float* w2out = (const float*)d_in[16];
  const float* w3a1  = (const float*)d_in[17];
  const float* b3a1  = (const float*)d_in[18];
  const float* w3a2  = (const float*)d_in[19];
  const float* b3a2  = (const float*)d_in[20];
  const float* w3mid = (const float*)d_in[21];
  const float* w3out = (const float*)d_in[22];

  // Workspace layout (bytes)
  char* ws = (char*)d_ws;
  size_t off = 0;
  auto take = [&](size_t bytes) {
    char* p = ws + off;
    off += (bytes + 255) & ~(size_t)255;
    return p;
  };
  int*   knn_idx = (int*)  take((size_t)BB * HWN * KNN * 4);
  float* finalb  = (float*)take((size_t)BB * C3 * HWN * 4);   // interp weighted sum
  float* f3i     = (float*)take((size_t)BB * C3 * HWN * 4);
  float* a2      = (float*)take((size_t)BB * C2 * HWN * 4);
  float* b2      = (float*)take((size_t)BB * C2 * HWN * 4);
  float* samp    = (float*)take((size_t)BB * C2 * NN * 4);
  float* f2s     = (float*)take((size_t)BB * C2 * NN * 4);
  float* a3      = (float*)take((size_t)BB * C3 * NN * 4);
  float* b3      = (float*)take((size_t)BB * C3 * NN * 4);
  float* pool2   = (float*)take((size_t)BB * C2 * 4);
  float* wsel2   = (float*)take((size_t)BB * C2 * 2 * 4);
  float* pool3   = (float*)take((size_t)BB * C3 * 4);
  float* wsel3   = (float*)take((size_t)BB * C3 * 2 * 4);
  (void)ws_size;

  float* out2d = (float*)d_out;                       // (B, C2, H, W)
  float* out3d = out2d + (size_t)BB * C2 * HWN;       // (B, C3, N)

  // ---- 3D -> 2D branch ----
  knn_kernel<<<BB * (HWN / 256), 256, 0, stream>>>(uv, knn_idx);
  interp_score_kernel<<<(BB * HWN + 255) / 256, 256, 0, stream>>>(
      uv, feat_3d, knn_idx, w_s1, b_s1, w_s2, b_s2, finalb);
  // f3i = lrelu(w_oc @ final + b_oc)
  gemm_pw<C3, HWN, 1><<<dim3(HWN / 16, C3 / 16, BB), 32, 0, stream>>>(
      w_oc, finalb, b_oc, f3i, C3);
  // a2 = lrelu(w2a1 @ feat2d + b2a1) ; b2 = lrelu(w2a2 @ f3i + b2a2)
  gemm_pw<C2, HWN, 1><<<dim3(HWN / 16, C2 / 16, BB), 32, 0, stream>>>(
      w2a1, feat_2d, b2a1, a2, C2);
  gemm_pw<C3, HWN, 1><<<dim3(HWN / 16, C2 / 16, BB), 32, 0, stream>>>(
      w2a2, f3i, b2a2, b2, C2);
  pool_mean<<<BB * C2, 256, 0, stream>>>(a2, b2, pool2, HWN);
  sk_fc<<<1, 256, 0, stream>>>(pool2, w2mid, w2out, wsel2, C2, C2 / 2);
  {
    const int total = BB * C2 * HWN;
    sk_combine<<<(total + 255) / 256, 256, 0, stream>>>(a2, b2, wsel2, out2d, HWN, total);
  }

  // ---- 2D -> 3D branch ----
  grid_sample_kernel<<<(BB * NN + 255) / 256, 256, 0, stream>>>(feat_2d, uv, samp);
  // f2s = lrelu(w_m3 @ samp + b_m3)
  gemm_pw<C2, NN, 1><<<dim3(NN / 16, C2 / 16, BB), 32, 0, stream>>>(
      w_m3, samp, b_m3, f2s, C2);
  // a3 = lrelu(w3a1 @ f2s + b3a1) ; b3 = lrelu(w3a2 @ feat3d + b3a2)
  gemm_pw<C2, NN, 1><<<dim3(NN / 16, C3 / 16, BB), 32, 0, stream>>>(
      w3a1, f2s, b3a1, a3, C3);
  gemm_pw<C3, NN, 1><<<dim3(NN / 16, C3 / 16, BB), 32, 0, stream>>>(
      w3a2, feat_3d, b3a2, b3, C3);
  pool_mean<<<BB * C3, 256, 0, stream>>>(a3, b3, pool3, NN);
  sk_fc<<<1, 256, 0, stream>>>(pool3, w3mid, w3out, wsel3, C3, C3 / 2);
  {
    const int total = BB * C3 * NN;
    sk_combine<<<(total + 255) / 256, 256, 0, stream>>>(a3, b3, wsel3, out3d, NN, total);
  }
}